// DAGNessLoss_6605659701675
// MI455X (gfx1250) — compile-verified
//
#include <hip/hip_runtime.h>
#include <cmath>

typedef __attribute__((ext_vector_type(2))) float v2f;
typedef __attribute__((ext_vector_type(8))) float v8f;

// One workgroup = 8 wave32s. Each thread strides the diagonal, accumulates
// exp(x^2) in f32, then each wave reduces its 32 partials with a single
// V_WMMA_F32_16X16X4_F32 (A = partials, B = ones -> D holds row sums).
__global__ __launch_bounds__(256) void diag_exp_trace_kernel(
    const float* __restrict__ W0, float* __restrict__ out, int N) {
  const int tid = threadIdx.x;
  const size_t stride = (size_t)N + 1;  // diagonal stride in elements

  float p = 0.0f;
#pragma unroll 8
  for (int i = tid; i < N; i += 256) {
    float x = W0[(size_t)i * stride];
    p += expf(x * x);
  }

  // ---- wave32 reduction via WMMA ----
  // A (16x4 f32): lane holds {p, 0} -> each row m of A sums to p_m + p_{m+16}.
  // B (4x16 f32): all ones.
  // D[m][n] = sum_k A[m][k]  (row sums, broadcast across N columns).
  v2f a; a.x = p;    a.y = 0.0f;
  v2f b; b.x = 1.0f; b.y = 1.0f;
  v8f c = {};
  c = __builtin_amdgcn_wmma_f32_16x16x4_f32(
      /*neg_a=*/false, a, /*neg_b=*/false, b,
      /*c_mod=*/(short)0, c, /*reuse_a=*/false, /*reuse_b=*/false);

  // In-lane sum of the 8 D VGPRs: lanes 0-15 now hold sum of rows 0..7,
  // lanes 16-31 hold sum of rows 8..15 (column N = lane%16, identical per column).
  float s = c[0] + c[1] + c[2] + c[3] + c[4] + c[5] + c[6] + c[7];
  float waveSum =
      __int_as_float(__builtin_amdgcn_readlane(__float_as_int(s), 0)) +
      __int_as_float(__builtin_amdgcn_readlane(__float_as_int(s), 16));

  // ---- combine 8 waves through LDS ----
  __shared__ float wsums[8];
  const int wave = tid >> 5;
  if ((tid & 31) == 0) wsums[wave] = waveSum;
  __syncthreads();

  if (tid == 0) {
    float tr = 0.0f;
#pragma unroll
    for (int w = 0; w < 8; ++w) tr += wsums[w];
    float d = tr - (float)N;
    out[0] = d * d;
  }
}

extern "C" void kernel_launch(void* const* d_in, const int* in_sizes, int n_in,
                              void* d_out, int out_size, void* d_ws, size_t ws_size,
                              hipStream_t stream) {
  (void)n_in; (void)out_size; (void)d_ws; (void)ws_size;
  const float* W0 = (const float*)d_in[0];
  float* out = (float*)d_out;
  const long long total = (long long)in_sizes[0];      // N*N
  const int N = (int)llround(sqrt((double)total));      // 8192 for the reference
  diag_exp_trace_kernel<<<1, 256, 0, stream>>>(W0, out, N);
}